// GCNLayer_23407571763910
// MI455X (gfx1250) — compile-verified
//
#include <hip/hip_runtime.h>

#define D_FEAT 128
#define THREADS 256

// ===========================================================================
// Fallback path: direct edge-parallel atomic scatter (used if ws too small).
// ===========================================================================
__global__ __launch_bounds__(THREADS) void gcn_zero_out(float4* __restrict__ out, int n4) {
    int i = blockIdx.x * blockDim.x + threadIdx.x;
    if (i < n4) out[i] = make_float4(0.f, 0.f, 0.f, 0.f);
}

__global__ __launch_bounds__(THREADS) void gcn_spmm_atomic(
    const int*   __restrict__ rows,
    const int*   __restrict__ cols,
    const float* __restrict__ vals,
    const float* __restrict__ embeds,
    float*       __restrict__ out,
    int nEdges)
{
    const int tid  = blockIdx.x * blockDim.x + threadIdx.x;
    const int lane = threadIdx.x & 31;

    int   r = 0, c = 0;
    float v = 0.0f;
    if (tid < nEdges) { r = rows[tid]; c = cols[tid]; v = vals[tid]; }

    const char* base = (const char*)(embeds + (size_t)c * D_FEAT);
    __builtin_prefetch(base +   0, 0, 1);
    __builtin_prefetch(base + 128, 0, 1);
    __builtin_prefetch(base + 256, 0, 1);
    __builtin_prefetch(base + 384, 0, 1);

#pragma unroll
    for (int j = 0; j < 32; ++j) {
        const int   cj = __builtin_amdgcn_readlane(c, j);
        const int   rj = __builtin_amdgcn_readlane(r, j);
        const float vj = __int_as_float(__builtin_amdgcn_readlane(__float_as_int(v), j));
        const float4 g = ((const float4*)(embeds + (size_t)cj * D_FEAT))[lane];
        float* o = out + (size_t)rj * D_FEAT + lane * 4;
        unsafeAtomicAdd(o + 0, vj * g.x);
        unsafeAtomicAdd(o + 1, vj * g.y);
        unsafeAtomicAdd(o + 2, vj * g.z);
        unsafeAtomicAdd(o + 3, vj * g.w);
    }
}

// ===========================================================================
// Fast path: device-built 32-padded CSR, then atomic-free row accumulation.
// ===========================================================================

// --- Phase 1a: zero the row-degree histogram ---
__global__ __launch_bounds__(THREADS) void gcn_zero_counts(int* __restrict__ counts, int n) {
    int i = blockIdx.x * blockDim.x + threadIdx.x;
    if (i < n) counts[i] = 0;
}

// --- Phase 1b: zero the (padded) pair buffer so pad slots read (col=0,val=0) ---
__global__ __launch_bounds__(THREADS) void gcn_zero_pairs(int2* __restrict__ pairs, int cap) {
    int i = blockIdx.x * blockDim.x + threadIdx.x;
    if (i < cap) pairs[i] = make_int2(0, 0);
}

// --- Phase 2: histogram of destination rows (native u32 atomics, no return) ---
__global__ __launch_bounds__(THREADS) void gcn_hist(
    const int* __restrict__ rows, int* __restrict__ counts, int nEdges) {
    int e = blockIdx.x * blockDim.x + threadIdx.x;
    if (e < nEdges) atomicAdd(&counts[rows[e]], 1);
}

// --- Phase 3: single-block exclusive scan of 32-PADDED degrees ---
__global__ __launch_bounds__(THREADS) void gcn_scan(
    const int* __restrict__ counts,
    int* __restrict__ offsets,   // n+1 entries (padded CSR row starts)
    int* __restrict__ cursor,    // n entries (scatter cursors)
    int n)
{
    __shared__ int partial[THREADS];
    const int t   = threadIdx.x;
    const int per = (n + THREADS - 1) / THREADS;
    const int beg = t * per;
    const int end = min(beg + per, n);

    int s = 0;
    for (int i = beg; i < end; ++i) s += (counts[i] + 31) & ~31;
    partial[t] = s;
    __syncthreads();

    if (t == 0) {
        int run = 0;
        for (int i = 0; i < THREADS; ++i) { int x = partial[i]; partial[i] = run; run += x; }
        offsets[n] = run;   // total padded edge slots
    }
    __syncthreads();

    int run = partial[t];
    for (int i = beg; i < end; ++i) {
        offsets[i] = run;
        cursor[i]  = run;
        run += (counts[i] + 31) & ~31;
    }
}

// --- Phase 4: scatter (col, val) pairs into padded-CSR order ---
__global__ __launch_bounds__(THREADS) void gcn_scatter(
    const int*   __restrict__ rows,
    const int*   __restrict__ cols,
    const float* __restrict__ vals,
    int*  __restrict__ cursor,
    int2* __restrict__ pairs,
    int nEdges)
{
    int e = blockIdx.x * blockDim.x + threadIdx.x;
    if (e < nEdges) {
        int r   = rows[e];
        int pos = atomicAdd(&cursor[r], 1);
        pairs[pos] = make_int2(cols[e], __float_as_int(vals[e]));
    }
}

// --- Phase 5: one wave32 per row; double-buffered, fully-unrolled, no atomics.
// Bucket length is a multiple of 32, so there is no tail path: every chunk is
// one coalesced global_load_b64 of 32 (col,val) pairs, 32 v_readlane
// broadcasts, 32 cooperative global_load_b128 gathers (512 contiguous
// bytes/edge), and 128 FMAs. One global_store_b128 per lane at the end.
__global__ __launch_bounds__(THREADS) void gcn_row_accum(
    const int*   __restrict__ offsets,
    const int2*  __restrict__ pairs,
    const float* __restrict__ embeds,
    float*       __restrict__ out,
    int nRows)
{
    const int gtid = blockIdx.x * blockDim.x + threadIdx.x;
    const int lane = threadIdx.x & 31;
    // Wave-uniform row: offsets[row], offsets[row+1] become scalar loads.
    const int row  = __builtin_amdgcn_readfirstlane(gtid >> 5);
    if (row >= nRows) return;

    const int beg = offsets[row];
    const int end = offsets[row + 1];   // (end - beg) % 32 == 0

    float4 acc = make_float4(0.f, 0.f, 0.f, 0.f);

    // Software pipeline: current chunk's pairs in p, next chunk prefetched.
    int2 p = make_int2(0, 0);
    if (beg < end) p = pairs[beg + lane];

    for (int k = beg; k < end; k += 32) {
        const int kn = k + 32;
        int2 pn = make_int2(0, 0);
        if (kn < end) pn = pairs[kn + lane];

        // Warm all 4 cachelines of this lane's embedding row
        // (global_prefetch_b8 on gfx1250).
        const char* pb = (const char*)(embeds + (size_t)p.x * D_FEAT);
        __builtin_prefetch(pb +   0, 0, 1);
        __builtin_prefetch(pb + 128, 0, 1);
        __builtin_prefetch(pb + 256, 0, 1);
        __builtin_prefetch(pb + 384, 0, 1);

#pragma unroll
        for (int j = 0; j < 32; ++j) {
            const int   cj = __builtin_amdgcn_readlane(p.x, j);
            const float vj = __int_as_float(__builtin_amdgcn_readlane(p.y, j));
            const float4 g = ((const float4*)(embeds + (size_t)cj * D_FEAT))[lane];
            acc.x = fmaf(vj, g.x, acc.x);
            acc.y = fmaf(vj, g.y, acc.y);
            acc.z = fmaf(vj, g.z, acc.z);
            acc.w = fmaf(vj, g.w, acc.w);
        }
        p = pn;
    }

    // One clean 16B store per lane: out[row][lane*4 .. lane*4+3].
    ((float4*)(out + (size_t)row * D_FEAT))[lane] = acc;
}

// ===========================================================================
// Host launcher.
// Inputs: adj_rows i32[E], adj_cols i32[E], adj_vals f32[E], embeds f32[N*128]
// ===========================================================================
extern "C" void kernel_launch(void* const* d_in, const int* in_sizes, int n_in,
                              void* d_out, int out_size, void* d_ws, size_t ws_size,
                              hipStream_t stream) {
    const int*   rows   = (const int*)  d_in[0];
    const int*   cols   = (const int*)  d_in[1];
    const float* vals   = (const float*)d_in[2];
    const float* embeds = (const float*)d_in[3];
    float*       out    = (float*)      d_out;

    const int nEdges = in_sizes[0];
    const int nRows  = out_size / D_FEAT;   // == N_NODES

    // Padded-CSR capacity upper bound: every row padded up by at most 31.
    const int pairCap = nEdges + 31 * nRows;

    // Workspace layout: offsets[n+1] | cursor[n] | counts[n] | pad | pairs[cap]
    size_t offB   = 0;
    size_t curB   = offB + (size_t)(nRows + 1) * sizeof(int);
    size_t cntB   = curB + (size_t)nRows * sizeof(int);
    size_t pairB  = (cntB + (size_t)nRows * sizeof(int) + 15) & ~(size_t)15;
    size_t needed = pairB + (size_t)pairCap * sizeof(int2);

    const int edgeBlocks = (nEdges + THREADS - 1) / THREADS;

    if (ws_size >= needed) {
        char* ws      = (char*)d_ws;
        int*  offsets = (int*) (ws + offB);
        int*  cursor  = (int*) (ws + curB);
        int*  counts  = (int*) (ws + cntB);
        int2* pairs   = (int2*)(ws + pairB);

        const int rowBlocks  = (nRows + THREADS - 1) / THREADS;
        const int pairBlocks = (pairCap + THREADS - 1) / THREADS;

        gcn_zero_counts<<<rowBlocks,  THREADS, 0, stream>>>(counts, nRows);
        gcn_zero_pairs <<<pairBlocks, THREADS, 0, stream>>>(pairs, pairCap);
        gcn_hist       <<<edgeBlocks, THREADS, 0, stream>>>(rows, counts, nEdges);
        gcn_scan       <<<1,          THREADS, 0, stream>>>(counts, offsets, cursor, nRows);
        gcn_scatter    <<<edgeBlocks, THREADS, 0, stream>>>(rows, cols, vals, cursor, pairs, nEdges);

        const int waveBlocks = ((nRows * 32) + THREADS - 1) / THREADS;
        gcn_row_accum<<<waveBlocks, THREADS, 0, stream>>>(offsets, pairs, embeds, out, nRows);
    } else {
        // Fallback: direct atomic scatter.
        const int n4 = out_size / 4;
        gcn_zero_out<<<(n4 + THREADS - 1) / THREADS, THREADS, 0, stream>>>((float4*)out, n4);
        gcn_spmm_atomic<<<edgeBlocks, THREADS, 0, stream>>>(rows, cols, vals, embeds, out, nEdges);
    }
}